// HeteroGNNDetoxifier_21912923144482
// MI455X (gfx1250) — compile-verified
//
#include <hip/hip_runtime.h>
#include <math.h>

typedef __attribute__((ext_vector_type(2))) float v2f;
typedef __attribute__((ext_vector_type(8))) float v8f;

#define HID 128
#define NH 8
#define HD 16
#define KCH 64      // k-chunk staged in LDS
#define LDSTR 68    // padded LDS row stride (floats): banks 4*lm mod 64, conflict-free

__device__ __forceinline__ int imin(int a, int b) { return a < b ? a : b; }

// ---------------------------------------------------------------------------
// Y[M,128] = X[M,K] @ W[K,128] + bias[128]     (fp32 WMMA 16x16x4)
// block = (32,8): 8 waves; wave w owns 16x16 tile (rows blockIdx.x*16, cols w*16).
// All waves share the same 16 X-rows -> stage 16xKCH tile in LDS once per chunk
// (cooperative float4 loads, ds_load_b64 A-fragments), prefetch next chunk.
// A frag (16x4 f32): lanes 0-15 K={k0,k0+1}, lanes 16-31 K={k0+2,k0+3}
// B frag (4x16 f32): same K split across lane halves, col = lane&15
// C/D (16x16 f32):   VGPR r -> row r (lanes 0-15) / row r+8 (lanes 16-31)
// ---------------------------------------------------------------------------
__global__ __launch_bounds__(256) void gemm_bias_k(
    const float* __restrict__ X, const float* __restrict__ W,
    const float* __restrict__ bias, float* __restrict__ Y,
    int M, int K) {
  __shared__ float atile[16 * LDSTR];
  const int lane = threadIdx.x;
  const int half = lane >> 4;
  const int lm   = lane & 15;
  const int tid  = threadIdx.y * 32 + lane;
  const int r0   = blockIdx.x * 16;
  const int c0   = threadIdx.y * 16;

  // cooperative-load mapping: thread -> (row 0..15, 4-float column group)
  const int lr = tid >> 4;
  const int lc = (tid & 15) << 2;
  const int arow = imin(r0 + lr, M - 1);          // clamp; garbage rows never stored
  const float* __restrict__ xrow = X + (size_t)arow * K;

  v8f acc = {};
  for (int kc = 0; kc < K; kc += KCH) {
    float4 xv = *(const float4*)(xrow + kc + lc);
    if (kc + KCH < K)                              // uniform branch
      __builtin_prefetch(xrow + kc + KCH + lc, 0, 1);
    __syncthreads();                               // prev chunk fully consumed
    *(float4*)(&atile[lr * LDSTR + lc]) = xv;
    __syncthreads();                               // tile visible to all waves
#pragma unroll
    for (int kk = 0; kk < KCH; kk += 4) {
      const int ka = kk + 2 * half;
      v2f a, b;
      a.x = atile[lm * LDSTR + ka];
      a.y = atile[lm * LDSTR + ka + 1];
      b.x = W[(size_t)(kc + ka) * HID + c0 + lm];
      b.y = W[(size_t)(kc + ka + 1) * HID + c0 + lm];
      acc = __builtin_amdgcn_wmma_f32_16x16x4_f32(false, a, false, b,
                                                  (short)0, acc, false, false);
    }
  }
#pragma unroll
  for (int i = 0; i < 8; ++i) {
    const int row = r0 + i + 8 * half;
    if (row < M) Y[(size_t)row * HID + c0 + lm] = acc[i] + bias[c0 + lm];
  }
}

// ---------------------------------------------------------------------------
// Per-head relation transform: Y[n, h*16+j] = sum_d X[n, h*16+d] * A[h][d][j]
// block = (32,8): wave w == head h; grid.x covers row tiles of 16. K=16 only.
// ---------------------------------------------------------------------------
__global__ __launch_bounds__(256) void rel_k(
    const float* __restrict__ X, const float* __restrict__ A,
    float* __restrict__ Y, int M) {
  const int lane = threadIdx.x;
  const int half = lane >> 4;
  const int lm   = lane & 15;
  const int h    = threadIdx.y;
  const int r0   = blockIdx.x * 16;
  const int ar   = imin(r0 + lm, M - 1);
  const float* __restrict__ xrow = X + (size_t)ar * HID + h * HD;
  const float* __restrict__ Wh   = A + h * (HD * HD);

  v8f acc = {};
#pragma unroll
  for (int k0 = 0; k0 < HD; k0 += 4) {
    const int ka = k0 + 2 * half;
    v2f a, b;
    a.x = xrow[ka];
    a.y = xrow[ka + 1];
    b.x = Wh[ka * HD + lm];
    b.y = Wh[(ka + 1) * HD + lm];
    acc = __builtin_amdgcn_wmma_f32_16x16x4_f32(false, a, false, b,
                                                (short)0, acc, false, false);
  }
#pragma unroll
  for (int i = 0; i < 8; ++i) {
    const int row = r0 + i + 8 * half;
    if (row < M) Y[(size_t)row * HID + h * HD + lm] = acc[i];
  }
}

// monotonic float <-> ordered-uint mapping (works for all finite floats)
__device__ __forceinline__ unsigned f2ord(float f) {
  unsigned b = __float_as_uint(f);
  return (b & 0x80000000u) ? ~b : (b | 0x80000000u);
}
__device__ __forceinline__ float ord2f(unsigned o) {
  unsigned b = (o & 0x80000000u) ? (o & 0x7fffffffu) : ~o;
  return __uint_as_float(b);
}

// pass 1: raw logits + running segment max (ordered-uint atomicMax)
__global__ __launch_bounds__(256) void alpha_k(
    const int* __restrict__ src, const int* __restrict__ dst, int E,
    const float* __restrict__ Q, const float* __restrict__ KREL,
    const float* __restrict__ prel, float* __restrict__ ALPHA,
    unsigned* __restrict__ Mb) {
  const int t = blockIdx.x * blockDim.x + threadIdx.x;
  if (t >= E * NH) return;
  const int e = t >> 3, h = t & 7;
  const int s = src[e], d = dst[e];
  const float4* __restrict__ qr = (const float4*)(Q + (size_t)d * HID + h * HD);
  const float4* __restrict__ kr = (const float4*)(KREL + (size_t)s * HID + h * HD);
  float acc = 0.f;
#pragma unroll
  for (int j = 0; j < 4; ++j) {
    float4 q4 = qr[j], k4 = kr[j];
    acc += q4.x * k4.x + q4.y * k4.y + q4.z * k4.z + q4.w * k4.w;
  }
  const float a = acc * prel[h] * 0.25f;  // 1/sqrt(D), D=16
  ALPHA[t] = a;
  atomicMax(&Mb[(size_t)d * NH + h], f2ord(a));
}

// pass 2: exp(a - max) + segment sum
__global__ __launch_bounds__(256) void expsum_k(
    const int* __restrict__ dst, int E, float* __restrict__ ALPHA,
    const unsigned* __restrict__ Mb, float* __restrict__ Sb) {
  const int t = blockIdx.x * blockDim.x + threadIdx.x;
  if (t >= E * NH) return;
  const int e = t >> 3, h = t & 7;
  const int d = dst[e];
  const float m  = ord2f(Mb[(size_t)d * NH + h]);
  const float ex = __expf(ALPHA[t] - m);
  ALPHA[t] = ex;
  unsafeAtomicAdd(&Sb[(size_t)d * NH + h], ex);
}

// pass 3: weighted scatter of relation-transformed values into OUT[dst]
__global__ __launch_bounds__(256) void scatter_k(
    const int* __restrict__ src, const int* __restrict__ dst, int E,
    const float* __restrict__ ALPHA, const float* __restrict__ Sb,
    const float* __restrict__ VREL, float* __restrict__ OUT) {
  const int t = blockIdx.x * blockDim.x + threadIdx.x;
  if (t >= E * NH) return;
  const int e = t >> 3, h = t & 7;
  const int s = src[e], d = dst[e];
  const float w = ALPHA[t] / (Sb[(size_t)d * NH + h] + 1e-16f);
  const float4* __restrict__ vr = (const float4*)(VREL + (size_t)s * HID + h * HD);
  float* __restrict__ o = OUT + (size_t)d * HID + h * HD;
#pragma unroll
  for (int j = 0; j < 4; ++j) {
    float4 v = vr[j];
    unsafeAtomicAdd(o + 4 * j + 0, w * v.x);
    unsafeAtomicAdd(o + 4 * j + 1, w * v.y);
    unsafeAtomicAdd(o + 4 * j + 2, w * v.z);
    unsafeAtomicAdd(o + 4 * j + 3, w * v.w);
  }
}

// tanh-approx GELU (jax.nn.gelu default, approximate=True)
__global__ __launch_bounds__(256) void gelu_k(
    const float* __restrict__ in, float* __restrict__ out, int total) {
  const int i = blockIdx.x * blockDim.x + threadIdx.x;
  if (i >= total) return;
  const float x = in[i];
  const float c = 0.7978845608028654f;
  const float t = tanhf(c * (x + 0.044715f * x * x * x));
  out[i] = 0.5f * x * (1.f + t);
}

// xs = sigmoid(skip[t]) * o + (1 - sigmoid(skip[t])) * xs
__global__ __launch_bounds__(256) void combine_k(
    float* __restrict__ XS, const float* __restrict__ O,
    const float* __restrict__ skip, int base, int n0, int n01, int total) {
  const int i = blockIdx.x * blockDim.x + threadIdx.x;
  if (i >= total) return;
  const int row = i >> 7;
  const int t = (row < n0) ? 0 : ((row < n01) ? 1 : 2);
  const float beta = 1.f / (1.f + __expf(-skip[base + t]));
  XS[i] = beta * O[i] + (1.f - beta) * XS[i];
}

// ---------------------------------------------------------------------------
extern "C" void kernel_launch(void* const* d_in, const int* in_sizes, int n_in,
                              void* d_out, int out_size, void* d_ws, size_t ws_size,
                              hipStream_t stream) {
  // inputs (setup_inputs order)
  const float* xin[3] = {(const float*)d_in[0], (const float*)d_in[1], (const float*)d_in[2]};
  const int* esrc[5]; const int* edst[5];
  int Ecnt[5];
  for (int e = 0; e < 5; ++e) {
    esrc[e] = (const int*)d_in[3 + 2 * e];
    edst[e] = (const int*)d_in[4 + 2 * e];
    Ecnt[e] = in_sizes[3 + 2 * e];
  }
  const float* Win[3] = {(const float*)d_in[13], (const float*)d_in[15], (const float*)d_in[17]};
  const float* bin[3] = {(const float*)d_in[14], (const float*)d_in[16], (const float*)d_in[18]};
  const float* Wk = (const float*)d_in[19]; const float* bk = (const float*)d_in[20];
  const float* Wq = (const float*)d_in[21]; const float* bq = (const float*)d_in[22];
  const float* Wv = (const float*)d_in[23]; const float* bv = (const float*)d_in[24];
  const float* Wa = (const float*)d_in[25]; const float* ba = (const float*)d_in[26];
  const float* skip  = (const float*)d_in[27];
  const float* a_rel = (const float*)d_in[28];
  const float* m_rel = (const float*)d_in[29];
  const float* p_rel = (const float*)d_in[30];

  const int Kin[3] = {512, 1024, 128};
  int ns[3];
  ns[0] = in_sizes[0] / 512; ns[1] = in_sizes[1] / 1024; ns[2] = in_sizes[2] / 128;
  const int NTOT = ns[0] + ns[1] + ns[2];
  const int off[3] = {0, ns[0], ns[0] + ns[1]};
  const int stp[5] = {2, 0, 0, 1, 1};
  const int dtp[5] = {0, 2, 1, 0, 1};

  int maxNs = ns[0] > ns[1] ? ns[0] : ns[1]; if (ns[2] > maxNs) maxNs = ns[2];
  int maxE = 0; for (int e = 0; e < 5; ++e) if (Ecnt[e] > maxE) maxE = Ecnt[e];

  // workspace layout (floats)
  float* XS   = (float*)d_ws;
  float* KB   = XS   + (size_t)NTOT * HID;
  float* QB   = KB   + (size_t)NTOT * HID;
  float* VB   = QB   + (size_t)NTOT * HID;
  float* OUTB = VB   + (size_t)NTOT * HID;
  float* KREL = OUTB + (size_t)NTOT * HID;
  float* VREL = KREL + (size_t)maxNs * HID;
  float* ALPH = VREL + (size_t)maxNs * HID;
  float* SB   = ALPH + (size_t)maxE * NH;
  unsigned* MB = (unsigned*)(SB + (size_t)maxNs * NH);

  const dim3 gblk(32, 8);

  // input projections: XS[t] = x[t] @ Win[t] + bin[t]
  for (int t = 0; t < 3; ++t)
    gemm_bias_k<<<dim3((ns[t] + 15) / 16), gblk, 0, stream>>>(
        xin[t], Win[t], bin[t], XS + (size_t)off[t] * HID, ns[t], Kin[t]);

  for (int l = 0; l < 2; ++l) {
    // K/Q/V projections per node type
    for (int t = 0; t < 3; ++t) {
      const size_t wo = (size_t)(l * 3 + t) * HID * HID;
      const size_t bo = (size_t)(l * 3 + t) * HID;
      const dim3 g((ns[t] + 15) / 16);
      gemm_bias_k<<<g, gblk, 0, stream>>>(XS + (size_t)off[t] * HID, Wk + wo, bk + bo,
                                          KB + (size_t)off[t] * HID, ns[t], HID);
      gemm_bias_k<<<g, gblk, 0, stream>>>(XS + (size_t)off[t] * HID, Wq + wo, bq + bo,
                                          QB + (size_t)off[t] * HID, ns[t], HID);
      gemm_bias_k<<<g, gblk, 0, stream>>>(XS + (size_t)off[t] * HID, Wv + wo, bv + bo,
                                          VB + (size_t)off[t] * HID, ns[t], HID);
    }

    hipMemsetAsync(OUTB, 0, (size_t)NTOT * HID * sizeof(float), stream);

    for (int e = 0; e < 5; ++e) {
      const int st = stp[e], dt = dtp[e];
      const int Ns = ns[st], Nd = ns[dt], E = Ecnt[e];
      const size_t ro = (size_t)(l * 5 + e) * NH * HD * HD;
      // relation-specific per-head transforms of K and V (source type)
      rel_k<<<dim3((Ns + 15) / 16), gblk, 0, stream>>>(KB + (size_t)off[st] * HID,
                                                       a_rel + ro, KREL, Ns);
      rel_k<<<dim3((Ns + 15) / 16), gblk, 0, stream>>>(VB + (size_t)off[st] * HID,
                                                       m_rel + ro, VREL, Ns);
      hipMemsetAsync(MB, 0, (size_t)Nd * NH * sizeof(unsigned), stream);
      hipMemsetAsync(SB, 0, (size_t)Nd * NH * sizeof(float), stream);
      const int nb = (E * NH + 255) / 256;
      alpha_k<<<nb, 256, 0, stream>>>(esrc[e], edst[e], E,
                                      QB + (size_t)off[dt] * HID, KREL,
                                      p_rel + (size_t)(l * 5 + e) * NH, ALPH, MB);
      expsum_k<<<nb, 256, 0, stream>>>(edst[e], E, ALPH, MB, SB);
      scatter_k<<<nb, 256, 0, stream>>>(esrc[e], edst[e], E, ALPH, SB, VREL,
                                        OUTB + (size_t)off[dt] * HID);
    }

    // o = gelu(out) @ Wa + ba ; xs = beta*o + (1-beta)*xs
    const int tot = NTOT * HID;
    gelu_k<<<(tot + 255) / 256, 256, 0, stream>>>(OUTB, KB, tot);   // reuse KB
    for (int t = 0; t < 3; ++t) {
      const size_t wo = (size_t)(l * 3 + t) * HID * HID;
      const size_t bo = (size_t)(l * 3 + t) * HID;
      gemm_bias_k<<<dim3((ns[t] + 15) / 16), gblk, 0, stream>>>(
          KB + (size_t)off[t] * HID, Wa + wo, ba + bo,
          VB + (size_t)off[t] * HID, ns[t], HID);                   // reuse VB as o
    }
    combine_k<<<(tot + 255) / 256, 256, 0, stream>>>(XS, VB, skip, l * 3,
                                                     ns[0], ns[0] + ns[1], tot);
  }

  hipMemcpyAsync(d_out, XS, (size_t)NTOT * HID * sizeof(float),
                 hipMemcpyDeviceToDevice, stream);
}